// HungarianMatcher_987842478259
// MI455X (gfx1250) — compile-verified
//
#include <hip/hip_runtime.h>

#define NH 7
#define BS 16
#define NQ 900
#define NT 100
#define NC 16
#define QSTRIDE 912            // padded query stride (multiple of 16)
#define NQT 57                 // ceil(900/16)
#define NTT 7                  // ceil(100/16)
#define NTILES (NQT * NTT)     // 399 tiles per batch

typedef __attribute__((ext_vector_type(2))) float v2f;
typedef __attribute__((ext_vector_type(8))) float v8f;

// ---------------------------------------------------------------------------
// Phase 1: fused cost matrix.  One wave = one 16x16 (q,t) tile of batch b.
// Label cost via f32 WMMA (A = sigmoid(logits), B = one-hot(labels)), exact.
// A 16x4 f32 layout: lane L holds row M=L%16; K={0,1} (lanes 0-15), {2,3} (16-31).
// B 4x16 f32 layout (symmetric): lane L holds col N=L%16; same K split.
// C/D 16x16 f32:  VGPR r, lane L -> (M = r + 8*(L>=16), N = L%16).
// ---------------------------------------------------------------------------
__global__ __launch_bounds__(256)
void hm_cost_kernel(const float* __restrict__ logits,   // (7,16,900,16)
                    const float* __restrict__ pboxes,   // (16,900,4)
                    const float* __restrict__ tboxes,   // (16,100,4)
                    const int*   __restrict__ tlabels,  // (7,16,100)
                    float* __restrict__ C)              // (16,100,QSTRIDE)
{
    const int b    = blockIdx.x;
    const int wave = threadIdx.x >> 5;
    const int lane = threadIdx.x & 31;
    const int tile = blockIdx.y * 8 + wave;
    if (tile >= NTILES) return;          // uniform per wave: EXEC stays full

    const int qt = tile / NTT;
    const int tt = tile % NTT;
    const int qbase = qt * 16;
    const int tbase = tt * 16;
    const int l16 = lane & 15;
    const int hi  = lane >> 4;           // 0: K lanes {0,1}, 1: K lanes {2,3}

    // this lane's target column (clamped for loads; OOB columns never stored)
    const int t  = tbase + l16;
    const int tc = (t < NT) ? t : (NT - 1);
    int lab[NH];
#pragma unroll
    for (int h = 0; h < NH; ++h)
        lab[h] = tlabels[((size_t)h * BS + b) * NT + tc];

    // this lane's A row (clamped; OOB rows never stored)
    const int qa  = qbase + l16;
    const int qac = (qa < NQ) ? qa : (NQ - 1);

    v8f acc = {};
#pragma unroll
    for (int step = 0; step < 28; ++step) {         // K = 7*16 = 112 = 28*4
        const int k0 = step * 4 + hi * 2;           // even; k0,k0+1 same head
        const int h  = k0 >> 4;
        const int c  = k0 & 15;
        const float* ap = logits + (((size_t)h * BS + b) * NQ + qac) * NC + c;
        v2f a, bm;
        a.x  = 1.0f / (1.0f + __expf(-ap[0]));
        a.y  = 1.0f / (1.0f + __expf(-ap[1]));
        bm.x = (lab[h] == c    ) ? 1.0f : 0.0f;
        bm.y = (lab[h] == c + 1) ? 1.0f : 0.0f;
        acc = __builtin_amdgcn_wmma_f32_16x16x4_f32(false, a, false, bm,
                                                    (short)0, acc, false, false);
    }

    // target box (shared by this lane's 8 outputs)
    const float* tb = tboxes + ((size_t)b * NT + tc) * 4;
    const float tcx = tb[0], tcy = tb[1], tw = tb[2], th = tb[3];
    const float tx0 = tcx - 0.5f * tw, ty0 = tcy - 0.5f * th;
    const float tx1 = tcx + 0.5f * tw, ty1 = tcy + 0.5f * th;
    const float area_t = (tx1 - tx0) * (ty1 - ty0);

#pragma unroll
    for (int r = 0; r < 8; ++r) {
        const int q = qbase + r + hi * 8;
        if (q >= NQ || t >= NT) continue;
        const float* pb = pboxes + ((size_t)b * NQ + q) * 4;
        const float pcx = pb[0], pcy = pb[1], pw = pb[2], ph = pb[3];
        const float l1 = fabsf(pcx - tcx) + fabsf(pcy - tcy)
                       + fabsf(pw - tw)   + fabsf(ph - th);
        const float px0 = pcx - 0.5f * pw, py0 = pcy - 0.5f * ph;
        const float px1 = pcx + 0.5f * pw, py1 = pcy + 0.5f * ph;
        const float area_p = (px1 - px0) * (py1 - py0);
        const float ix0 = fmaxf(px0, tx0), iy0 = fmaxf(py0, ty0);
        const float ix1 = fminf(px1, tx1), iy1 = fminf(py1, ty1);
        const float iw = fmaxf(ix1 - ix0, 0.0f), ih = fmaxf(iy1 - iy0, 0.0f);
        const float inter = iw * ih;
        const float uni   = area_p + area_t - inter;
        const float iou   = inter / uni;
        const float ex0 = fminf(px0, tx0), ey0 = fminf(py0, ty0);
        const float ex1 = fmaxf(px1, tx1), ey1 = fmaxf(py1, ty1);
        const float ew = fmaxf(ex1 - ex0, 0.0f), eh = fmaxf(ey1 - ey0, 0.0f);
        const float enc  = ew * eh;
        const float giou = iou - (enc - uni) / enc;
        // C = 5*L1 + 2*(-labels) + 2*(-giou); stored transposed: [b][t][q]
        C[((size_t)b * NT + t) * QSTRIDE + q] = 5.0f * l1 - 2.0f * acc[r] - 2.0f * giou;
    }
}

// ---------------------------------------------------------------------------
// Phase 2: Jonker-Volgenant shortest augmenting path, one workgroup per batch.
// Rows = 100 targets, cols = 900 queries.  float64 duals (matches numpy).
// Argmin tie-break: smallest column index (== np.argmin first-hit semantics).
// ---------------------------------------------------------------------------
__global__ __launch_bounds__(1024)
void hm_lsa_kernel(const float* __restrict__ C, float* __restrict__ out)
{
    const int b   = blockIdx.x;
    const int tid = threadIdx.x;
    const float* Cb = C + (size_t)b * NT * QSTRIDE;

    __shared__ double s_minv[NQ + 1];
    __shared__ double s_v[NQ + 1];
    __shared__ double s_u[NT + 1];
    __shared__ int    s_way[NQ + 1];
    __shared__ int    s_p[NQ + 1];
    __shared__ int    s_used[NQ + 1];
    __shared__ double s_rv[1024];
    __shared__ int    s_ri[1024];
    __shared__ int    s_t2p[NT];
    __shared__ int    s_j0, s_done;

    const double DINF = __builtin_inf();

    if (tid <= NQ) { s_v[tid] = 0.0; s_p[tid] = 0; s_way[tid] = 0; }
    if (tid <= NT) s_u[tid] = 0.0;
    __syncthreads();

    for (int i = 1; i <= NT; ++i) {
        if (tid == 0) { s_p[0] = i; s_j0 = 0; s_done = 0; }
        if (tid <= NQ) { s_minv[tid] = DINF; s_used[tid] = 0; }
        __syncthreads();

        for (int iter = 0; iter < NQ + 2; ++iter) {
            const int j0 = s_j0;
            if (tid == 0) s_used[j0] = 1;
            __syncthreads();
            const int    i0  = s_p[j0];
            const double ui0 = s_u[i0];

            if (tid >= 1 && tid <= NQ && !s_used[tid]) {
                double cur = (double)Cb[(size_t)(i0 - 1) * QSTRIDE + (tid - 1)]
                             - ui0 - s_v[tid];
                if (cur < s_minv[tid]) { s_minv[tid] = cur; s_way[tid] = j0; }
            }
            __syncthreads();

            // tie-broken argmin over free columns
            s_rv[tid] = (tid >= 1 && tid <= NQ && !s_used[tid]) ? s_minv[tid] : DINF;
            s_ri[tid] = tid;
            __syncthreads();
            for (int s = 512; s > 0; s >>= 1) {
                if (tid < s) {
                    const double ov = s_rv[tid + s];
                    const int    oi = s_ri[tid + s];
                    if (ov < s_rv[tid] || (ov == s_rv[tid] && oi < s_ri[tid])) {
                        s_rv[tid] = ov; s_ri[tid] = oi;
                    }
                }
                __syncthreads();
            }
            const double delta = s_rv[0];
            const int    j1    = s_ri[0];

            if (tid <= NQ) {
                if (s_used[tid]) {
                    s_v[tid] -= delta;
                    s_u[s_p[tid]] += delta;   // rows on tree are distinct
                } else {
                    s_minv[tid] -= delta;
                }
            }
            __syncthreads();
            if (tid == 0) { s_j0 = j1; s_done = (s_p[j1] == 0); }
            __syncthreads();
            if (s_done) break;
        }

        if (tid == 0) {                      // backtrack augmenting path
            int j0 = s_j0;
            while (j0) { const int j1 = s_way[j0]; s_p[j0] = s_p[j1]; j0 = j1; }
        }
        __syncthreads();
    }

    // row2col, then stable-sort by assigned column (values are distinct)
    if (tid >= 1 && tid <= NQ && s_p[tid] > 0) s_t2p[s_p[tid] - 1] = tid - 1;
    __syncthreads();
    if (tid < NT) {
        const int val = s_t2p[tid];
        int rank = 0;
        for (int j = 0; j < NT; ++j) rank += (s_t2p[j] < val) ? 1 : 0;
        out[(size_t)b * 2 * NT + rank]      = (float)val;   // out[b,0,:]
        out[(size_t)b * 2 * NT + NT + rank] = (float)tid;   // out[b,1,:]
    }
}

extern "C" void kernel_launch(void* const* d_in, const int* in_sizes, int n_in,
                              void* d_out, int out_size, void* d_ws, size_t ws_size,
                              hipStream_t stream)
{
    (void)in_sizes; (void)n_in; (void)out_size; (void)ws_size;
    const float* logits  = (const float*)d_in[0];   // (7,16,900,16) f32
    const float* pboxes  = (const float*)d_in[1];   // (16,900,4)    f32
    const float* tboxes  = (const float*)d_in[2];   // (16,100,4)    f32
    const int*   tlabels = (const int*)d_in[3];     // (7,16,100)    i32
    float* Cmat = (float*)d_ws;                     // 16*100*912*4 = 5.84 MB
    float* out  = (float*)d_out;                    // 16*2*100 floats

    dim3 gridA(BS, (NTILES + 7) / 8);               // 8 waves/block, 1 tile/wave
    hm_cost_kernel<<<gridA, 256, 0, stream>>>(logits, pboxes, tboxes, tlabels, Cmat);
    hm_lsa_kernel<<<BS, 1024, 0, stream>>>(Cmat, out);
}